// MessagePassing_7937099563205
// MI455X (gfx1250) — compile-verified
//
#include <hip/hip_runtime.h>

// Segment-sum scatter: out[src[e], k] += edge_attrs_flat[k*E + e]
// E = 4,000,000 edges, F = 16 features, N = 100,000 nodes.
//
// Memory-bound: 256MB vals (stream once, NT) + 16MB idx (NT) + 6.4MB output
// accumulated via L2-resident global_atomic_add_f32. HBM floor ~12us @ 23.3TB/s.
// CDNA5 paths: async global->LDS DMA (global_load_async_to_lds_b128 / ASYNCcnt)
// double-buffered pipeline, with TH_LOAD_NT so the 272MB one-shot stream does
// not evict the 6.4MB accumulator from the 192MB L2. WMMA does not apply:
// scatter-add over 100K random segments has no dense-matmul structure (a
// one-hot WMMA formulation merges ~nothing per 16-slot tile and still needs
// the same atomics).

#define FEATS      16
#define TILE_EDGES 256           // edges per tile; E % 256 == 0 (15625 tiles)
#define BLOCK      256           // 8 wave32 per block
#define NBLOCKS    1024

__device__ __forceinline__ void async_copy16_nt(const float* gsrc, unsigned lds_off) {
    // Per-lane 16B global -> LDS DMA; tracked by ASYNCcnt (not LOADcnt).
    // TH_LOAD_NT: values are consumed exactly once -> keep them low-priority
    // in cache so the output accumulator owns L2.
    asm volatile("global_load_async_to_lds_b128 %0, %1, off th:TH_LOAD_NT"
                 :: "v"(lds_off), "v"(gsrc)
                 : "memory");
}

__global__ void __launch_bounds__(BLOCK)
zero_kernel(float* __restrict__ out, int n) {
    int i = blockIdx.x * BLOCK + threadIdx.x;
    if (i < n) out[i] = 0.0f;   // RT store: accumulator lines want L2 residency
}

__global__ void __launch_bounds__(BLOCK)
scatter_sum_kernel(const float* __restrict__ vals,  // flat, indexed as (F, E)
                   const int*   __restrict__ src,   // attr_idx row 0, E ints
                   float*       __restrict__ out,   // (N, F) f32
                   long long    E_ll)
{
    const size_t E = (size_t)E_ll;
    const int nTiles = (int)(E / TILE_EDGES);

    // Double-buffered staging: [buf][feature][edge-in-tile], 2 x 16KB.
    __shared__ float lds[2][FEATS * TILE_EDGES];

    const int tid    = threadIdx.x;
    const int stride = gridDim.x;

    // Stage one 256-edge tile (16KB) into LDS buffer `buf`.
    // 1024 float4 units; each thread issues 4 async B128 DMAs.
    // Lanes of a wave cover 32 consecutive float4s of one feature row
    // -> each DMA instruction reads 512B contiguous global memory.
    auto stage = [&](int buf, int t) {
        const size_t base = (size_t)t * TILE_EDGES;
#pragma unroll
        for (int p = 0; p < 4; ++p) {
            const int u = tid + p * BLOCK;   // 0..1023
            const int k = u >> 6;            // feature row 0..15
            const int j = u & 63;            // float4 index within row
            const float* g = vals + (size_t)k * E + base + 4 * (size_t)j;
            const unsigned l =
                (unsigned)(uintptr_t)&lds[buf][k * TILE_EDGES + 4 * j];
            async_copy16_nt(g, l);
        }
    };

    int tile = blockIdx.x;
    if (tile < nTiles) stage(0, tile);

    int cur = 0;
    for (int t = tile; t < nTiles; t += stride) {
        // Issue the node-index load early (NT: read exactly once) so its
        // latency overlaps the DMA wait.
        const int node = __builtin_nontemporal_load(src + (size_t)t * TILE_EDGES + tid);

        const int tn = t + stride;
        if (tn < nTiles) {
            stage(cur ^ 1, tn);                             // prefetch next tile
            asm volatile("s_wait_asynccnt 4" ::: "memory"); // current tile done;
                                                            // next 4 DMAs in flight
        } else {
            asm volatile("s_wait_asynccnt 0" ::: "memory");
        }
        __syncthreads();   // all waves' staging of buf[cur] complete

        // Consume: 16 conflict-free ds_load_b32 + 16 f32 atomics into the
        // L2-resident (6.4MB) output row of this edge's node.
        float* orow = out + (size_t)node * FEATS;
#pragma unroll
        for (int k = 0; k < FEATS; ++k) {
            const float v = lds[cur][k * TILE_EDGES + tid];
            unsafeAtomicAdd(orow + k, v);   // global_atomic_add_f32, no return
        }

        __syncthreads();   // buf[cur] fully read before it is restaged
        cur ^= 1;
    }
}

extern "C" void kernel_launch(void* const* d_in, const int* in_sizes, int n_in,
                              void* d_out, int out_size, void* d_ws, size_t ws_size,
                              hipStream_t stream) {
    const float* edge_attrs = (const float*)d_in[0];  // E*F floats, row-major (E,F)
    const int*   attr_idx   = (const int*)d_in[1];    // (2,E) int32; row 0 = src
    float*       out        = (float*)d_out;          // N*F floats

    const long long E = (long long)in_sizes[0] / FEATS;   // 4,000,000

    // d_out is poisoned by the harness -> zero before accumulation.
    const int zgrid = (out_size + BLOCK - 1) / BLOCK;
    zero_kernel<<<zgrid, BLOCK, 0, stream>>>(out, out_size);

    scatter_sum_kernel<<<NBLOCKS, BLOCK, 0, stream>>>(edge_attrs, attr_idx, out, E);
}